// CrossEntropy_88931592831632
// MI455X (gfx1250) — compile-verified
//
#include <hip/hip_runtime.h>
#include <math.h>

// ---------------------------------------------------------------------------
// Hypergraph BCE loss (pairwise + tuplewise) for gfx1250 / MI455X.
//
// 16-node tiles. A = 16 node rows staged in padded LDS (fragments preloaded
// to VGPRs), B = 16 gathered neighbor rows per slot. Per slot: chained
// V_WMMA_F32_16X16X4_F32 over K=128 (two independent accumulator chains),
// read the diagonal of the 16x16 result. 6 waves/block: wave0 = positives,
// waves1..5 = the 5 negative groups of 8 (matches reshape(N,5,8).mean(2)).
//
// Gathers use GLOBAL_LOAD_ASYNC_TO_LDS_B128 (ASYNCcnt) with wave-private
// double-buffered B tiles: slot s+1's gather overlaps slot s's WMMA, with
// s_wait_asynccnt as the only sync (no per-slot block barriers). Falls back
// to a synchronous ds-staged path if the async builtins are unavailable.
//
// Per-block partials -> d_ws; deterministic fixed-order reduction -> d_out.
// ---------------------------------------------------------------------------

typedef __attribute__((ext_vector_type(2))) float v2f;
typedef __attribute__((ext_vector_type(8))) float v8f;

#define TILE_M   16
#define D_DIM    128
#define ROW_PAD  132            // 128 + 4 words: LDS bank-conflict-free frags
#define NWAVES   6              // one wave per group of 8 slots
#define BLOCK    (NWAVES * 32)  // wave32
#define KSTEPS   (D_DIM / 4)    // 32 chained v_wmma_f32_16x16x4_f32

#if __has_builtin(__builtin_amdgcn_global_load_async_to_lds_b128) && \
    __has_builtin(__builtin_amdgcn_s_wait_asynccnt)
#define USE_ASYNC 1
#else
#define USE_ASYNC 0
#endif

#if USE_ASYNC
typedef int v4i_t __attribute__((vector_size(16)));
typedef __attribute__((address_space(1))) char  as1_char;
typedef __attribute__((address_space(3))) char  as3_char;
typedef __attribute__((address_space(1))) v4i_t as1_v4i;
typedef __attribute__((address_space(3))) v4i_t as3_v4i;
#define WAIT_ASYNC(n) __builtin_amdgcn_s_wait_asynccnt(n)

// 256B (half row) memory -> LDS, 16 x async b128, no VGPR round trip.
__device__ __forceinline__ void async_copy_256B(const float* gsrc, float* ldst) {
  as1_char* g = (as1_char*)gsrc;
  as3_char* l = (as3_char*)ldst;
#pragma unroll
  for (int k = 0; k < 16; ++k)
    __builtin_amdgcn_global_load_async_to_lds_b128(
        (as1_v4i*)(g + 16 * k), (as3_v4i*)(l + 16 * k), 0, 0);
}
#endif

__device__ __forceinline__ float sigmoid_f(float x) {
  return 1.0f / (1.0f + expf(-x));
}
__device__ __forceinline__ float softplus_f(float x) {
  return logf(1.0f + expf(x));   // args here are in (0,1): no overflow risk
}

// neighbor row index for (wave, slot) of node n
__device__ __forceinline__ int slot_index(const int* __restrict__ pos_idx,
                                          const int* __restrict__ neg_idx,
                                          int n, int wave, int s8) {
  return (wave == 0) ? pos_idx[n * 8 + s8]
                     : neg_idx[n * 40 + (wave - 1) * 8 + s8];
}

// D = A(16x128) x B(128x16) via 32 f32 WMMAs in two independent chains;
// return this lane's diagonal element (lane<8 or lane>=24 own one).
__device__ __forceinline__ float wmma_diag(const float* bbuf, const v2f* afrag,
                                           int mn, int g, int diag_reg) {
  v8f c0 = {}, c1 = {};
#pragma unroll
  for (int s = 0; s < KSTEPS; s += 2) {
    v2f b0 = *(const v2f*)(&bbuf[mn * ROW_PAD + 4 * s + 2 * g]);
    v2f b1 = *(const v2f*)(&bbuf[mn * ROW_PAD + 4 * (s + 1) + 2 * g]);
    // 8 args: (neg_a, A, neg_b, B, c_mod, C, reuse_a, reuse_b)
    c0 = __builtin_amdgcn_wmma_f32_16x16x4_f32(false, afrag[s], false, b0,
                                               (short)0, c0, false, false);
    c1 = __builtin_amdgcn_wmma_f32_16x16x4_f32(false, afrag[s + 1], false, b1,
                                               (short)0, c1, false, false);
  }
  float d = 0.0f;
#pragma unroll
  for (int r = 0; r < 8; ++r)
    if (r == diag_reg) d = c0[r] + c1[r];
  return d;
}

__global__ __launch_bounds__(BLOCK, 1)
void hg_loss_tile_kernel(const float* __restrict__ x,
                         const int* __restrict__ pos_idx,
                         const int* __restrict__ neg_idx,
                         float* __restrict__ ws, int N) {
  __shared__ float Abuf[TILE_M * ROW_PAD];
  __shared__ float Bbuf[NWAVES][2][TILE_M * ROW_PAD];  // wave-private, 2x buf
  __shared__ float pairLDS[NWAVES * TILE_M];
  __shared__ float sigLDS [NWAVES * TILE_M];
  __shared__ float resLDS [2 * TILE_M];

  const int tid  = threadIdx.x;
  const int wave = tid >> 5;
  const int lane = tid & 31;
  const int tile = blockIdx.x;
  const int n0   = tile * TILE_M;

  // gather assignment: 2 lanes per 512B neighbor row
  const int grow  = lane >> 1;
  const int ghalf = lane & 1;
  int gn = n0 + grow; if (gn >= N) gn = N - 1;

#if USE_ASYNC
  // kick off slot 0 gather; overlaps A staging below
  {
    int nbr = slot_index(pos_idx, neg_idx, gn, wave, 0);
    async_copy_256B(x + (size_t)nbr * D_DIM + ghalf * 64,
                    &Bbuf[wave][0][grow * ROW_PAD + ghalf * 64]);
  }
#endif

  // ---- stage A tile (16 node rows) into LDS, float4 granularity -----------
  for (int t = tid; t < TILE_M * (D_DIM / 4); t += BLOCK) {
    int row = t >> 5;               // 32 float4 segments per row
    int seg = t & 31;
    int n = n0 + row; if (n >= N) n = N - 1;
    const float4 v = ((const float4*)(x + (size_t)n * D_DIM))[seg];
    *(float4*)(&Abuf[row * ROW_PAD + seg * 4]) = v;
  }
  __syncthreads();

  // ---- preload this lane's A fragments for all 32 k-steps (VGPR-resident) -
  // 32-bit A 16x4 layout: lane = (k>=2)*16 + m ; vgpr = k&1  => k = 2g + v
  const int g  = lane >> 4;
  const int mn = lane & 15;         // M for A frags / N for B frags
  v2f afrag[KSTEPS];
#pragma unroll
  for (int s = 0; s < KSTEPS; ++s)
    afrag[s] = *(const v2f*)(&Abuf[mn * ROW_PAD + 4 * s + 2 * g]);

  // diagonal ownership in the 16x16 f32 C/D layout:
  //   lane<8  -> node lane    in c[lane] ; lane>=24 -> node lane-16 in c[lane-24]
  const bool diag_active = (lane < 8) || (lane >= 24);
  const int  diag_node   = lane & 15;
  const int  diag_reg    = lane & 7;

  float pair_acc = 0.0f;   // softplus(sig) (- sig for positives)
  float sig_acc  = 0.0f;   // sum of sigmoids (group-mean numerator)

#if USE_ASYNC
  for (int s8 = 0; s8 < 8; ++s8) {
    const float* cur = &Bbuf[wave][s8 & 1][0];
    if (s8 < 7) {
      // prefetch slot s8+1 into the other buffer, then wait for slot s8
      int nbr = slot_index(pos_idx, neg_idx, gn, wave, s8 + 1);
      async_copy_256B(x + (size_t)nbr * D_DIM + ghalf * 64,
                      &Bbuf[wave][(s8 + 1) & 1][grow * ROW_PAD + ghalf * 64]);
      WAIT_ASYNC(16);                 // 16 in flight = next batch only
    } else {
      WAIT_ASYNC(0);
    }
    asm volatile("" ::: "memory");    // keep ds reads below the wait

    float d = wmma_diag(cur, afrag, mn, g, diag_reg);
    if (diag_active) {
      float sg = sigmoid_f(d);
      float sp = softplus_f(sg);
      pair_acc += (wave == 0) ? (sp - sg) : sp;
      sig_acc  += sg;
    }
  }
#else
  for (int s8 = 0; s8 < 8; ++s8) {
    {  // synchronous gather through VGPRs into this wave's buffer 0
      int nbr = slot_index(pos_idx, neg_idx, gn, wave, s8);
      const float4* src = (const float4*)(x + (size_t)nbr * D_DIM) + ghalf * 16;
      float4* dst = (float4*)(&Bbuf[wave][0][grow * ROW_PAD + ghalf * 64]);
#pragma unroll
      for (int k = 0; k < 16; ++k) dst[k] = src[k];
    }
    __syncthreads();                  // uniform: every wave runs 8 slots

    float d = wmma_diag(&Bbuf[wave][0][0], afrag, mn, g, diag_reg);
    if (diag_active) {
      float sg = sigmoid_f(d);
      float sp = softplus_f(sg);
      pair_acc += (wave == 0) ? (sp - sg) : sp;
      sig_acc  += sg;
    }
    __syncthreads();                  // protect buffer reuse next iteration
  }
#endif

  if (diag_active) {
    pairLDS[wave * TILE_M + diag_node] = pair_acc;
    sigLDS [wave * TILE_M + diag_node] = sig_acc;
  }
  __syncthreads();

  // ---- per-node finalize (first 16 threads), then serial 16-sum -----------
  if (tid < TILE_M) {
    float pair = 0.0f;
#pragma unroll
    for (int w = 0; w < NWAVES; ++w) pair += pairLDS[w * TILE_M + tid];
    float per_node_pair = pair * (1.0f / 48.0f);

    float pos_t = sigLDS[0 * TILE_M + tid] * 0.125f;
    float tup = softplus_f(pos_t) - pos_t;
#pragma unroll
    for (int k = 1; k < NWAVES; ++k) {
      float nt = sigLDS[k * TILE_M + tid] * 0.125f;
      tup += softplus_f(nt);
    }
    float per_node_tuple = tup * (1.0f / 6.0f);

    bool valid = (n0 + tid) < N;
    resLDS[tid]          = valid ? per_node_pair  : 0.0f;
    resLDS[TILE_M + tid] = valid ? per_node_tuple : 0.0f;
  }
  __syncthreads();

  if (tid == 0) {
    float a = 0.0f, b = 0.0f;
#pragma unroll
    for (int i = 0; i < TILE_M; ++i) { a += resLDS[i]; b += resLDS[TILE_M + i]; }
    ws[2 * tile]     = a;   // pairwise partial
    ws[2 * tile + 1] = b;   // tuplewise partial
  }
}

// Deterministic fixed-order reduction of per-tile partials -> scalar output.
__global__ __launch_bounds__(256, 1)
void hg_loss_reduce_kernel(const float* __restrict__ ws, int ntiles,
                           const int* __restrict__ hyper,
                           float* __restrict__ out, int N) {
  __shared__ float sa[256];
  __shared__ float sb[256];
  const int tid = threadIdx.x;
  float a = 0.0f, b = 0.0f;
  for (int i = tid; i < ntiles; i += 256) { a += ws[2 * i]; b += ws[2 * i + 1]; }
  sa[tid] = a; sb[tid] = b;
  __syncthreads();
#pragma unroll
  for (int off = 128; off > 0; off >>= 1) {
    if (tid < off) { sa[tid] += sa[tid + off]; sb[tid] += sb[tid + off]; }
    __syncthreads();
  }
  if (tid == 0) {
    float denom = (float)(N - hyper[0]);
    out[0] = sa[0] / (float)N + sb[0] / denom;
  }
}

extern "C" void kernel_launch(void* const* d_in, const int* in_sizes, int n_in,
                              void* d_out, int out_size, void* d_ws, size_t ws_size,
                              hipStream_t stream) {
  const float* x       = (const float*)d_in[0];   // [N,128] fp32
  const int*   pos_idx = (const int*)  d_in[1];   // [N,8]
  const int*   neg_idx = (const int*)  d_in[2];   // [N,40]
  const int*   hyper   = (const int*)  d_in[3];   // scalar

  const int N      = in_sizes[0] / D_DIM;
  const int ntiles = (N + TILE_M - 1) / TILE_M;
  float* ws = (float*)d_ws;                       // 2 floats per tile

  hg_loss_tile_kernel<<<ntiles, BLOCK, 0, stream>>>(x, pos_idx, neg_idx, ws, N);
  hg_loss_reduce_kernel<<<1, 256, 0, stream>>>(ws, ntiles, hyper,
                                               (float*)d_out, N);
}